// Conv2dModulation_60808146976824
// MI455X (gfx1250) — compile-verified
//
#include <hip/hip_runtime.h>
#include <hip/hip_bf16.h>

typedef __attribute__((ext_vector_type(2))) float v2f;
typedef __attribute__((ext_vector_type(4))) float v4f;
typedef __attribute__((ext_vector_type(8))) float v8f;

#define B_   16
#define CIN  64
#define COUT 64
#define TAPS 9
#define H_   256
#define W_   256
#define HW   (H_ * W_)

// Paired-ci LDS layouts (ci pair p = ci>>1, parity = ci&1):
//   Ws[p][cout][2]  stride padded 128->160 dwords  (160 % 64 == 32: the two
//                   K-pair lane groups hit disjoint LDS bank halves)
//   Xs[p][row][col][2] stride padded 396->416 dwords (416 % 64 == 32)
#define WS_PSTRIDE 160
#define WS_TOTAL2  (32 * WS_PSTRIDE)   // 5120 floats / 20 KB
#define XS_PSTRIDE 416
#define XS_TOTAL2  (32 * XS_PSTRIDE)   // 13312 floats / 53.2 KB

// ---------------------------------------------------------------------------
// Kernel 1: modulate + demodulate weights.
//   out layout: wmod[b][tap][ci/2][cout][2] with WS_PSTRIDE padding, so the
//   conv kernel's A-operand staging is a contiguous b128 copy and the
//   A fragment (W[k][m], W[k+1][m]) is one aligned ds_load_b64.
// ---------------------------------------------------------------------------
__global__ __launch_bounds__(64)
void modulate_demod_kernel(const float* __restrict__ weight,
                           const float* __restrict__ y,
                           float* __restrict__ wmod) {
    const int co = blockIdx.x;   // 0..63
    const int bb = blockIdx.y;   // 0..15
    const int ci = threadIdx.x;  // 0..63

    const float kaiming = 0.0416666666666666667f;  // (cin*k*k)^-0.5 = 1/24
    const float ym = y[bb * CIN + ci] * kaiming;

    const float* wp = weight + (co * CIN + ci) * TAPS;
    float v[TAPS];
    float ss = 0.f;
#pragma unroll
    for (int t = 0; t < TAPS; ++t) {
        v[t] = wp[t] * ym;
        ss += v[t] * v[t];
    }

    __shared__ float red[64];
    red[ci] = ss;
    __syncthreads();
#pragma unroll
    for (int s = 32; s > 0; s >>= 1) {
        if (ci < s) red[ci] += red[ci + s];
        __syncthreads();
    }
    float sum = red[0] + 1e-8f;
    float d = rsqrtf(sum);
    d = d * (1.5f - 0.5f * sum * d * d);  // Newton step -> ~fp32 accurate

#pragma unroll
    for (int t = 0; t < TAPS; ++t) {
        wmod[((bb * TAPS + t) * 32 + (ci >> 1)) * WS_PSTRIDE + co * 2 + (ci & 1)]
            = v[t] * d;
    }
}

// ---------------------------------------------------------------------------
// Kernel 2: implicit-GEMM grouped conv via V_WMMA_F32_16X16X4_F32.
//   Block = (sample, image row, 64-px x tile). X halo tile is staged with
//   GLOBAL_LOAD_ASYNC_TO_LDS_B32 (ASYNCcnt) with ds_store zeros on the SAME
//   padding border; weights staged per tap as b128 copies.
// ---------------------------------------------------------------------------
__global__ __launch_bounds__(256)
void modconv_wmma_kernel(const float* __restrict__ X,
                         const float* __restrict__ wmod,
                         float* __restrict__ out) {
    extern __shared__ float smem[];
    float* Xs = smem;                  // [32 pairs][3][66][2] (+pad)
    float* Ws = smem + XS_TOTAL2;      // [32 pairs][64 cout][2] (+pad)

    const int tid  = threadIdx.x;
    const int lane = tid & 31;
    const int wave = tid >> 5;

    const int xBase = blockIdx.x * 64;  // 0,64,128,192
    const int yRow  = blockIdx.y;       // 0..255
    const int bIdx  = blockIdx.z;       // 0..15

    // -------- async-stage X halo tile (rows y-1..y+1, cols x-1..x+64) ------
    {
        const int hBase = yRow - 1;
        const int wBase = xBase - 1;
        const float* Xb = X + bIdx * CIN * HW;
        for (int idx = tid; idx < CIN * 198; idx += 256) {
            const int ci  = idx / 198;
            const int rem = idx - ci * 198;
            const int row = rem / 66;
            const int col = rem - row * 66;
            const int lidx = (ci >> 1) * XS_PSTRIDE + row * 132 + col * 2 + (ci & 1);
            const int gy = hBase + row;
            const int gx = wBase + col;
            if ((unsigned)gy < (unsigned)H_ && (unsigned)gx < (unsigned)W_) {
                const float* gp = Xb + ci * HW + gy * W_ + gx;
                const unsigned ldsb = (unsigned)lidx * 4u;  // no static LDS -> raw offset
                asm volatile("global_load_async_to_lds_b32 %0, %1, off"
                             :: "v"(ldsb), "v"(gp) : "memory");
            } else {
                Xs[lidx] = 0.f;  // SAME zero padding via ds_store
            }
        }
        asm volatile("s_wait_asynccnt 0x0" ::: "memory");
    }

    // -------- wave tiling ---------------------------------------------------
    const int mBase = (wave & 3) * 16;   // cout tile base
    const int nHalf = (wave >> 2) * 32;  // pixel half of the 64-px tile
    const int m     = lane & 15;         // M index (== N index for B/C/D)
    const int kg    = lane >> 4;         // K pair-group per ISA operand layout

    v8f acc0 = {};  // pixels nHalf + 0..15
    v8f acc1 = {};  // pixels nHalf + 16..31

    const float* Wg = wmod + bIdx * TAPS * WS_TOTAL2;
    const float* wsBase = Ws + kg * WS_PSTRIDE + (mBase + m) * 2;

    for (int tap = 0; tap < TAPS; ++tap) {
        const int dy = tap / 3;
        const int dx = tap - dy * 3;

        __syncthreads();  // previous tap's Ws consumers done (also fences Xs)
        for (int i = tid * 4; i < WS_TOTAL2; i += 1024)
            *(v4f*)&Ws[i] = *(const v4f*)&Wg[tap * WS_TOTAL2 + i];
        __syncthreads();

        const float* xsBase =
            Xs + kg * XS_PSTRIDE + dy * 132 + (nHalf + m + dx) * 2;

#pragma unroll
        for (int k0 = 0; k0 < CIN; k0 += 4) {
            const int p2 = k0 >> 1;  // ci-pair advance (2 pairs per K=4 step)
            // A fragment: W[k][mBase+m], W[k+1][mBase+m]  (one ds_load_b64)
            v2f a = *(const v2f*)(wsBase + p2 * WS_PSTRIDE);
            // B fragments: X[k][px], X[k+1][px]           (one ds_load_b64 each)
            v2f b0 = *(const v2f*)(xsBase + p2 * XS_PSTRIDE);
            v2f b1 = *(const v2f*)(xsBase + p2 * XS_PSTRIDE + 32);

            acc0 = __builtin_amdgcn_wmma_f32_16x16x4_f32(
                false, a, false, b0, (short)0, acc0, false, false);
            acc1 = __builtin_amdgcn_wmma_f32_16x16x4_f32(
                false, a, false, b1, (short)0, acc1, false, false);
        }
    }

    // -------- store: D layout = VGPR r -> M=r (lanes 0-15) / M=r+8 (16-31) --
    const int coBase = mBase + ((lane >= 16) ? 8 : 0);
    const int pxBase = xBase + nHalf + m;
    float* outB = out + (bIdx * COUT) * HW + yRow * W_;
#pragma unroll
    for (int r = 0; r < 8; ++r) {
        float* orow = outB + (coBase + r) * HW;
        orow[pxBase]      = acc0[r];
        orow[pxBase + 16] = acc1[r];
    }
}

// ---------------------------------------------------------------------------
extern "C" void kernel_launch(void* const* d_in, const int* in_sizes, int n_in,
                              void* d_out, int out_size, void* d_ws, size_t ws_size,
                              hipStream_t stream) {
    const float* X      = (const float*)d_in[0];  // (16,64,256,256)
    const float* y      = (const float*)d_in[1];  // (16,64)
    const float* weight = (const float*)d_in[2];  // (64,64,3,3)
    float* out  = (float*)d_out;                  // (16,64,256,256)
    float* wmod = (float*)d_ws;                   // 16*9*5120 floats ~ 2.95 MB

    dim3 g1(COUT, B_);
    modulate_demod_kernel<<<g1, 64, 0, stream>>>(weight, y, wmod);

    dim3 g2(W_ / 64, H_, B_);
    const size_t shmem = (size_t)(XS_TOTAL2 + WS_TOTAL2) * sizeof(float); // 73728 B
    modconv_wmma_kernel<<<g2, 256, shmem, stream>>>(X, wmod, out);
}